// LabelSmoothingCrossEntropyWithSuperLoss_12953621365467
// MI455X (gfx1250) — compile-verified
//
#include <hip/hip_runtime.h>
#include <hip/hip_bf16.h>
#include <cstdint>

#define GRID 2048
#define TPB  256
#define WPB  8          // waves per block (wave32)
#define CDIM 330

constexpr float kEps = 0.1f;
constexpr float kLam = 0.25f;
constexpr float kTau = 5.7990926544605255f;        // ln(330)
constexpr float kE   = 2.7182818284590452f;
constexpr float kNegTwoOverE = -0.73575888234288464f;

typedef __attribute__((ext_vector_type(2))) float v2f;
typedef __attribute__((ext_vector_type(8))) float v8f;

// ---- async global -> LDS staging of one row-pair (2640 B = 165 x b128) ----
__device__ __forceinline__ void stage_pair(const float* __restrict__ g, float* l, int lane)
{
    uint64_t ga = (uint64_t)(uintptr_t)g + (uint64_t)(lane * 16);
    uint32_t la = (uint32_t)(uintptr_t)l + (uint32_t)(lane * 16);   // low 32 bits of generic shared ptr == LDS offset
    asm volatile("global_load_async_to_lds_b128 %0, %1, off"             :: "v"(la), "v"(ga) : "memory");
    asm volatile("global_load_async_to_lds_b128 %0, %1, off offset:512"  :: "v"(la), "v"(ga) : "memory");
    asm volatile("global_load_async_to_lds_b128 %0, %1, off offset:1024" :: "v"(la), "v"(ga) : "memory");
    asm volatile("global_load_async_to_lds_b128 %0, %1, off offset:1536" :: "v"(la), "v"(ga) : "memory");
    asm volatile("global_load_async_to_lds_b128 %0, %1, off offset:2048" :: "v"(la), "v"(ga) : "memory");
    if (lane < 5)
        asm volatile("global_load_async_to_lds_b128 %0, %1, off offset:2560" :: "v"(la), "v"(ga) : "memory");
}

__device__ __forceinline__ void process_row(const float* __restrict__ rp, long row,
                                            const int* __restrict__ tgt, int lane,
                                            float& accLS, float& accSL)
{
    float xs[10];
#pragma unroll
    for (int k = 0; k < 5; ++k) {
        const float2 v = *reinterpret_cast<const float2*>(rp + (2 * lane + 64 * k));
        xs[2 * k] = v.x; xs[2 * k + 1] = v.y;
    }
    const bool  tv   = lane < 10;                    // cols 320..329
    const float tail = tv ? rp[320 + lane] : 0.0f;

    float mx = tv ? tail : -3.0e38f;
#pragma unroll
    for (int i = 0; i < 10; ++i) mx = fmaxf(mx, xs[i]);
#pragma unroll
    for (int off = 16; off > 0; off >>= 1) mx = fmaxf(mx, __shfl_xor(mx, off, 32));

    float se = tv ? __expf(tail - mx) : 0.0f;
    float sr = tv ? tail : 0.0f;
#pragma unroll
    for (int i = 0; i < 10; ++i) { se += __expf(xs[i] - mx); sr += xs[i]; }
#pragma unroll
    for (int off = 16; off > 0; off >>= 1) {
        se += __shfl_xor(se, off, 32);
        sr += __shfl_xor(sr, off, 32);
    }

    const int   t   = tgt[row];
    const float xt  = rp[t];                          // gather from LDS copy
    const float lse = mx + __logf(se);
    const float li  = lse - xt;                       // per-sample NLL
    const float lsi = (float)CDIM * lse - sr;         // -sum(log_preds)

    // Lambert W0: reference init + 10 Halley steps
    const float y  = 0.5f * fmaxf(kNegTwoOverE, (li - kTau) / kLam);
    const float pp = sqrtf(fmaxf(2.0f * (kE * y + 1.0f), 0.0f));
    float w = (y < 0.0f) ? (-1.0f + pp * (1.0f - pp * (1.0f / 3.0f)))
                         : log1pf(fmaxf(y, 0.0f));
#pragma unroll
    for (int it = 0; it < 10; ++it) {
        const float ew = __expf(w);
        const float f  = w * ew - y;
        float den = ew * (w + 1.0f) - (w + 2.0f) * f / (2.0f * w + 2.0f);
        den = (den == 0.0f) ? 1e-30f : den;
        w -= f / den;
    }
    const float sigma = __expf(-w);                   // log(sigma) = -w
    const float sup   = (li - kTau) * sigma + kLam * w * w;

    accLS += lsi;
    accSL += sup;
}

__global__ __launch_bounds__(TPB) void sl_main(const float* __restrict__ out,
                                               const int* __restrict__ tgt,
                                               float* __restrict__ ws,
                                               int nPairs, int totalWaves)
{
    __shared__ __align__(16) float smem[WPB][2][2 * CDIM];   // per-wave double buffer
    __shared__ float pls[WPB], psl[WPB];

    const int lane = threadIdx.x & 31;
    const int wid  = threadIdx.x >> 5;
    const int gw   = blockIdx.x * WPB + wid;

    float accLS = 0.0f, accSL = 0.0f;

    int p = gw;
    int bufsel = 0;
    if (p < nPairs) stage_pair(out + (long)p * (2 * CDIM), &smem[wid][0][0], lane);

    for (; p < nPairs; p += totalWaves) {
        const int pn = p + totalWaves;
        if (pn < nPairs) {
            stage_pair(out + (long)pn * (2 * CDIM), &smem[wid][bufsel ^ 1][0], lane);
            asm volatile("s_wait_asynccnt 6" ::: "memory");   // oldest group (6 ops) done
        } else {
            asm volatile("s_wait_asynccnt 0" ::: "memory");
        }
        const float* cur = &smem[wid][bufsel][0];
#pragma unroll
        for (int r = 0; r < 2; ++r)
            process_row(cur + r * CDIM, (long)p * 2 + r, tgt, lane, accLS, accSL);
        bufsel ^= 1;
    }

    if (lane == 0) { pls[wid] = accLS; psl[wid] = accSL; }
    __syncthreads();
    if (threadIdx.x == 0) {
        float a = 0.0f, b = 0.0f;
#pragma unroll
        for (int i = 0; i < WPB; ++i) { a += pls[i]; b += psl[i]; }
        ws[blockIdx.x]        = a;   // label-smoothing partials
        ws[GRID + blockIdx.x] = b;   // superloss partials
    }
}

// ---- final reduction of 2*GRID partials via V_WMMA_F32_16X16X4_F32 ----
__global__ __launch_bounds__(32) void sl_finalize(const float* __restrict__ ws,
                                                  float* __restrict__ out, float Bf)
{
    const int lane = threadIdx.x;
    const float sA = kEps / ((float)CDIM * Bf);       // coeff for ls partials
    const float sB = (1.0f - kEps) / Bf;              // coeff for superloss partials
    const int m    = lane & 15;
    const int koff = (lane < 16) ? 0 : 2;             // f32 16x4 A-matrix lane layout

    v8f acc = {0.f, 0.f, 0.f, 0.f, 0.f, 0.f, 0.f, 0.f};
    v2f bm  = {1.0f, 1.0f};                           // all-ones B => row sums

    for (int c = 0; c < (2 * GRID) / 64; ++c) {       // 64 chunks x 64 values
        const int i0 = c * 64 + m * 4 + koff;
        const float s0 = (i0     < GRID) ? sA : sB;
        const float s1 = (i0 + 1 < GRID) ? sA : sB;
        v2f a = { ws[i0] * s0, ws[i0 + 1] * s1 };
        acc = __builtin_amdgcn_wmma_f32_16x16x4_f32(false, a, false, bm,
                                                    (short)0, acc, false, false);
    }
    float s = 0.0f;
#pragma unroll
    for (int i = 0; i < 8; ++i) s += acc[i];          // lanes 0-15: rows 0-7; lanes 16-31: rows 8-15
    const float tot = __shfl(s, 0, 32) + __shfl(s, 16, 32);
    if (lane == 0) out[0] = tot;
}

extern "C" void kernel_launch(void* const* d_in, const int* in_sizes, int n_in,
                              void* d_out, int out_size, void* d_ws, size_t ws_size,
                              hipStream_t stream)
{
    const float* outp = (const float*)d_in[0];
    const int*   tgt  = (const int*)d_in[1];
    const int B       = in_sizes[1];
    const int nPairs  = B / 2;
    const int totalWaves = GRID * WPB;
    float* ws = (float*)d_ws;

    sl_main<<<dim3(GRID), dim3(TPB), 0, stream>>>(outp, tgt, ws, nPairs, totalWaves);
    sl_finalize<<<dim3(1), dim3(32), 0, stream>>>(ws, (float*)d_out, (float)B);
}